// GCNLayer_81758997447374
// MI455X (gfx1250) — compile-verified
//
#include <hip/hip_runtime.h>

typedef __attribute__((ext_vector_type(2))) float v2f;
typedef __attribute__((ext_vector_type(8))) float v8f;

#define C_DIM 128
#define LDA   132   // padded LDS row stride (floats): bank = (4*row + k) % 64, conflict-free
#define MTILE 64    // rows per block: one B fragment feeds 4 WMMAs (4 M-subtiles)

// -------------------------------------------------------------------------
// Fused GEMM: H[n, 0:128] = X@W0, H[n,128:256] = X@W1, H[n,256:384] = X@root
// Block = 64 rows x 384 cols; 8 waves; each wave walks 3 column tiles.
// Per tile: preload all 32 B fragments (one load clause + single wait),
// then 32 K-steps x 4 V_WMMA_F32_16X16X4_F32 with zero memory stalls.
// -------------------------------------------------------------------------
__global__ __launch_bounds__(256)
void splineconv_gemm_kernel(const float* __restrict__ X,
                            const float* __restrict__ W,     // [2,C,C] row-major
                            const float* __restrict__ root,  // [C,C]
                            float* __restrict__ H,           // [N, 3C]
                            int N)
{
    __shared__ float As[MTILE * LDA];
    const int tid  = threadIdx.x;
    const int wave = tid >> 5;
    const int lane = tid & 31;
    const int row0 = blockIdx.x * MTILE;

    // cooperative load of the 64x128 A tile (float4, coalesced, zero-fill OOB)
    for (int i = tid; i < MTILE * (C_DIM / 4); i += 256) {
        const int r  = i >> 5;          // 32 float4s per row
        const int c4 = i & 31;
        float4 v = make_float4(0.f, 0.f, 0.f, 0.f);
        const int gr = row0 + r;
        if (gr < N) v = ((const float4*)(X + (size_t)gr * C_DIM))[c4];
        float* dst = &As[r * LDA + c4 * 4];
        dst[0] = v.x; dst[1] = v.y; dst[2] = v.z; dst[3] = v.w;
    }
    __syncthreads();

    const int hi  = lane >> 4;   // 0: lanes 0-15, 1: lanes 16-31
    const int l16 = lane & 15;

    for (int t = wave; t < 24; t += 8) {
        const int gcol = t * 16;
        const float* Bbase;
        if (gcol < C_DIM)          Bbase = W + gcol;
        else if (gcol < 2 * C_DIM) Bbase = W + (size_t)C_DIM * C_DIM + (gcol - C_DIM);
        else                       Bbase = root + (gcol - 2 * C_DIM);

        // fixed per-lane bases; all offsets below are compile-time immediates
        const float* Bl = Bbase + (size_t)(2 * hi) * C_DIM + l16;
        const float* Al = &As[l16 * LDA + 2 * hi];

        // ---- preload ALL B fragments for this column tile (independent
        //      loads -> one clause, one s_wait_loadcnt) ----
        v2f bf[32];
        #pragma unroll
        for (int k4 = 0; k4 < 32; ++k4) {
            bf[k4].x = Bl[(size_t)(4 * k4) * C_DIM];
            bf[k4].y = Bl[(size_t)(4 * k4 + 1) * C_DIM];
        }

        // ---- pure-WMMA K loop: no memory ops between matrix ops ----
        v8f acc[4] = {{}, {}, {}, {}};
        #pragma unroll
        for (int k4 = 0; k4 < 32; ++k4) {
            const int k0 = 4 * k4;
            v2f a0, a1, a2, a3;
            a0.x = Al[k0];              a0.y = Al[k0 + 1];
            a1.x = Al[16 * LDA + k0];   a1.y = Al[16 * LDA + k0 + 1];
            a2.x = Al[32 * LDA + k0];   a2.y = Al[32 * LDA + k0 + 1];
            a3.x = Al[48 * LDA + k0];   a3.y = Al[48 * LDA + k0 + 1];
            acc[0] = __builtin_amdgcn_wmma_f32_16x16x4_f32(
                         false, a0, false, bf[k4], (short)0, acc[0], false, false);
            acc[1] = __builtin_amdgcn_wmma_f32_16x16x4_f32(
                         false, a1, false, bf[k4], (short)0, acc[1], false, false);
            acc[2] = __builtin_amdgcn_wmma_f32_16x16x4_f32(
                         false, a2, false, bf[k4], (short)0, acc[2], false, false);
            acc[3] = __builtin_amdgcn_wmma_f32_16x16x4_f32(
                         false, a3, false, bf[k4], (short)0, acc[3], false, false);
        }

        // C/D layout: VGPR r -> M = r + 8*hi, N = l16
        #pragma unroll
        for (int s = 0; s < 4; ++s) {
            const int rbase = row0 + s * 16;
            float* Hp = H + (size_t)rbase * (3 * C_DIM) + gcol;
            #pragma unroll
            for (int r = 0; r < 8; ++r) {
                const int m = r + 8 * hi;
                if (rbase + m < N)
                    Hp[(size_t)m * (3 * C_DIM) + l16] = acc[s][r];
            }
        }
    }
}

// -------------------------------------------------------------------------
// Per-edge blend + scatter-add. One wave per edge: 32 lanes x float4 = 128 ch.
// H (76.8 MB) and agg (25.6 MB) are L2-resident on the 192 MB L2, so the
// random gathers and fp32 atomics run at L2 speed, not HBM.
// -------------------------------------------------------------------------
__global__ __launch_bounds__(256)
void scatter_kernel(const int* __restrict__ ei,   // [2,E]
                    const float* __restrict__ u,  // [E]
                    const float* __restrict__ H,  // [N,3C]
                    float* __restrict__ agg,      // [N,C]
                    float* __restrict__ deg,      // [N]
                    int E)
{
    const int e    = blockIdx.x * 8 + (threadIdx.x >> 5);
    const int lane = threadIdx.x & 31;
    if (e >= E) return;

    const int   s  = ei[e];
    const int   d  = ei[E + e];
    const float uu = u[e];
    const float w0 = 1.0f - uu;

    const float4* h0 = (const float4*)(H + (size_t)s * 3 * C_DIM);
    const float4* h1 = (const float4*)(H + (size_t)s * 3 * C_DIM + C_DIM);
    const float4 a = h0[lane];
    const float4 b = h1[lane];
    const float4 m = make_float4(w0 * a.x + uu * b.x, w0 * a.y + uu * b.y,
                                 w0 * a.z + uu * b.z, w0 * a.w + uu * b.w);

    float* ap = agg + (size_t)d * C_DIM + lane * 4;
    atomicAdd(ap + 0, m.x);
    atomicAdd(ap + 1, m.y);
    atomicAdd(ap + 2, m.z);
    atomicAdd(ap + 3, m.w);
    if (lane == 0) atomicAdd(deg + d, 1.0f);
}

// -------------------------------------------------------------------------
// out[n] = agg[n]/max(deg[n],1) + H[n, 256:384] + bias
// -------------------------------------------------------------------------
__global__ __launch_bounds__(256)
void combine_kernel(const float* __restrict__ agg,
                    const float* __restrict__ deg,
                    const float* __restrict__ H,
                    const float* __restrict__ bias,
                    float* __restrict__ out,
                    int N)
{
    const int n    = blockIdx.x * 8 + (threadIdx.x >> 5);
    const int lane = threadIdx.x & 31;
    if (n >= N) return;

    const float inv = 1.0f / fmaxf(deg[n], 1.0f);
    const float4 a  = ((const float4*)(agg + (size_t)n * C_DIM))[lane];
    const float4 r  = ((const float4*)(H + (size_t)n * 3 * C_DIM + 2 * C_DIM))[lane];
    const float4 bb = ((const float4*)bias)[lane];
    const float4 o  = make_float4(a.x * inv + r.x + bb.x,
                                  a.y * inv + r.y + bb.y,
                                  a.z * inv + r.z + bb.z,
                                  a.w * inv + r.w + bb.w);
    ((float4*)(out + (size_t)n * C_DIM))[lane] = o;
}

__global__ __launch_bounds__(256)
void zero_kernel(float* __restrict__ p, size_t n)
{
    const size_t i = (size_t)blockIdx.x * blockDim.x + threadIdx.x;
    if (i < n) p[i] = 0.0f;
}

// -------------------------------------------------------------------------
extern "C" void kernel_launch(void* const* d_in, const int* in_sizes, int n_in,
                              void* d_out, int out_size, void* d_ws, size_t ws_size,
                              hipStream_t stream)
{
    const float* x  = (const float*)d_in[0];
    const int*   ei = (const int*)d_in[1];
    const float* ea = (const float*)d_in[2];
    const float* W1 = (const float*)d_in[3];
    const float* r1 = (const float*)d_in[4];
    const float* b1 = (const float*)d_in[5];
    const float* W2 = (const float*)d_in[6];
    const float* r2 = (const float*)d_in[7];
    const float* b2 = (const float*)d_in[8];
    float* out = (float*)d_out;

    const int N = in_sizes[0] / C_DIM;
    const int E = in_sizes[1] / 2;

    float* ws   = (float*)d_ws;
    float* Hbuf = ws;                            // N * 384
    float* agg  = Hbuf + (size_t)N * 3 * C_DIM;  // N * 128
    float* deg  = agg  + (size_t)N * C_DIM;      // N      (contiguous with agg)
    float* hmid = deg  + N;                      // N * 128

    const int mblocks = (N + MTILE - 1) / MTILE;
    const int eblocks = (E + 7) / 8;
    const int nblocks = (N + 7) / 8;
    const size_t zn   = (size_t)N * (C_DIM + 1); // agg + deg in one pass
    const int zblocks = (int)((zn + 255) / 256);

    // ---- Layer 1 ----
    zero_kernel<<<zblocks, 256, 0, stream>>>(agg, zn);
    splineconv_gemm_kernel<<<mblocks, 256, 0, stream>>>(x, W1, r1, Hbuf, N);
    scatter_kernel<<<eblocks, 256, 0, stream>>>(ei, ea, Hbuf, agg, deg, E);
    combine_kernel<<<nblocks, 256, 0, stream>>>(agg, deg, Hbuf, b1, hmid, N);

    // ---- Layer 2 ----
    zero_kernel<<<zblocks, 256, 0, stream>>>(agg, zn);
    splineconv_gemm_kernel<<<mblocks, 256, 0, stream>>>(hmid, W2, r2, Hbuf, N);
    scatter_kernel<<<eblocks, 256, 0, stream>>>(ei, ea, Hbuf, agg, deg, E);
    combine_kernel<<<nblocks, 256, 0, stream>>>(agg, deg, Hbuf, b2, out, N);
}